// IntraAttention_29463475651504
// MI455X (gfx1250) — compile-verified
//
#include <hip/hip_runtime.h>
#include <cstdint>
#include <cstddef>

// Problem constants (match reference: B=128, N=1024, D=2048)
#define BB 128
#define NN 1024
#define DD 2048

typedef __attribute__((ext_vector_type(2))) float v2f;
typedef __attribute__((ext_vector_type(8))) float v8f;

// ---------------------------------------------------------------------------
// Kernel 1: q = h_t @ W_attn   (128 x 2048) = (128 x 2048) @ (2048 x 2048)
// fp32 WMMA: V_WMMA_F32_16X16X4_F32, one wave per 16x16 output tile.
//
// Fragment layouts (CDNA5 ISA 7.12.2, wave32):
//   A (16x4 f32):  lane L in [0,16): M=L, v0=K0, v1=K1 ; lane L in [16,32): M=L-16, v0=K2, v1=K3
//   B (4x16 f32):  lane L: N=L&15 ; lanes<16 hold rows K0(v0)/K1(v1), lanes>=16 hold K2(v0)/K3(v1)
//   C/D (16x16):   lane L, vgpr r: M = r + 8*(L>>4), N = L&15
// ---------------------------------------------------------------------------
__global__ __launch_bounds__(128) void qgemm_wmma_f32(
    const float* __restrict__ ht,   // [B, D]
    const float* __restrict__ W,    // [D, D]
    float* __restrict__ q)          // [B, D]
{
    const int lane  = threadIdx.x & 31;
    const int wave  = threadIdx.x >> 5;
    const int wtile = blockIdx.x * (blockDim.x >> 5) + wave;   // 0..1023

    const int mtile = wtile & 7;         // 128/16 = 8 row tiles
    const int ntile = wtile >> 3;        // 2048/16 = 128 col tiles

    const int half  = lane >> 4;         // 0: lanes 0-15, 1: lanes 16-31
    const int mrow  = (mtile << 4) + (lane & 15);
    const int ncol  = (ntile << 4) + (lane & 15);
    const int koff  = half << 1;         // K offset 0 or 2 within the 4-wide slab

    const float* __restrict__ arow = ht + (size_t)mrow * DD + koff;   // contiguous float2
    const float* __restrict__ bcol = W  + (size_t)koff * DD + ncol;   // stride-D column pair

    v8f acc = {};

    #pragma unroll 8
    for (int k = 0; k < DD; k += 4) {
        v2f a;
        {   // aligned 8-byte load of A pair (K=koff, koff+1) for this lane's row
            const v2f* p = reinterpret_cast<const v2f*>(arow + k);
            a = *p;
        }
        v2f b;
        b.x = bcol[(size_t)k * DD];
        b.y = bcol[(size_t)(k + 1) * DD];

        // D = A(16x4) * B(4x16) + C ; 8 args: neg_a, A, neg_b, B, c_mod, C, reuse_a, reuse_b
        acc = __builtin_amdgcn_wmma_f32_16x16x4_f32(
            false, a, false, b, (short)0, acc, false, false);
    }

    const int mbase = (mtile << 4) + (half << 3);
    #pragma unroll
    for (int r = 0; r < 8; ++r) {
        q[(size_t)(mbase + r) * DD + ncol] = acc[r];
    }
}

// ---------------------------------------------------------------------------
// Kernel 2: zero the c_t slab of d_out (harness poisons it with 0xAA).
// ---------------------------------------------------------------------------
__global__ __launch_bounds__(256) void zero_ct(float* __restrict__ ct, int n)
{
    int i = blockIdx.x * blockDim.x + threadIdx.x;
    if (i < n) ct[i] = 0.0f;
}

// ---------------------------------------------------------------------------
// Kernel 3: single fused pass over h.
//   e[b,n]      = q[b] . h[b,n]
//   exp_e       = exp(e)
//   ee          = exp_e / past_e
//   alpha       = ee / past_ee
//   new_past_e  = past_e + exp_e
//   new_past_ee = past_ee + ee
//   c_t[b]     += alpha * h[b,n]          (h row reused from registers)
//
// One block (256 thr = 8 waves) per (batch, chunk of 128 rows); thread t owns
// d in [8t, 8t+8): q fragment and c accumulator are register-resident. Each
// h row is read exactly once (two coalesced float4 loads per thread).
// ---------------------------------------------------------------------------
#define NCHUNK 8
#define ROWS   (NN / NCHUNK)   // 128 rows per block

__global__ __launch_bounds__(256) void fused_attn_pass(
    const float* __restrict__ h,        // [B, N, D]
    const float* __restrict__ q,        // [B, D]
    const float* __restrict__ past_e,   // [B, N]
    const float* __restrict__ past_ee,  // [B, N]
    float* __restrict__ alpha_out,      // [B, N]
    float* __restrict__ ct_out,         // [B, D] (pre-zeroed, atomically accumulated)
    float* __restrict__ npe_out,        // [B, N]
    float* __restrict__ npee_out)       // [B, N]
{
    const int tid  = threadIdx.x;
    const int lane = tid & 31;
    const int wid  = tid >> 5;

    const int b     = blockIdx.x / NCHUNK;
    const int chunk = blockIdx.x % NCHUNK;
    const int n0    = chunk * ROWS;

    const int d0 = tid * 8;             // 256 threads * 8 = 2048 = D

    __shared__ float red[8];            // per-wave partial dots
    __shared__ float alpha_s;

    // Register-resident q fragment and c accumulator for this thread's 8 d's.
    float qreg[8];
    float creg[8];
    {
        const float* __restrict__ qp = q + (size_t)b * DD + d0;
        #pragma unroll
        for (int j = 0; j < 8; ++j) { qreg[j] = qp[j]; creg[j] = 0.0f; }
    }

    const float* __restrict__ hbase = h + ((size_t)b * NN + n0) * DD + d0;

    for (int r = 0; r < ROWS; ++r) {
        const float* __restrict__ hrow = hbase + (size_t)r * DD;
        const float4 hv0 = *reinterpret_cast<const float4*>(hrow);
        const float4 hv1 = *reinterpret_cast<const float4*>(hrow + 4);

        // partial dot over this thread's 8 elements
        float p = hv0.x * qreg[0] + hv0.y * qreg[1] + hv0.z * qreg[2] + hv0.w * qreg[3]
                + hv1.x * qreg[4] + hv1.y * qreg[5] + hv1.z * qreg[6] + hv1.w * qreg[7];

        // wave32 tree reduction
        p += __shfl_down(p, 16, 32);
        p += __shfl_down(p, 8, 32);
        p += __shfl_down(p, 4, 32);
        p += __shfl_down(p, 2, 32);
        p += __shfl_down(p, 1, 32);
        if (lane == 0) red[wid] = p;
        __syncthreads();

        if (tid == 0) {
            float e = red[0] + red[1] + red[2] + red[3]
                    + red[4] + red[5] + red[6] + red[7];
            const size_t bn = (size_t)b * NN + (n0 + r);
            const float pe  = past_e[bn];
            const float pee = past_ee[bn];
            const float ex  = expf(e);
            const float ee  = ex / pe;
            const float al  = ee / pee;
            alpha_out[bn] = al;
            npe_out[bn]   = pe + ex;
            npee_out[bn]  = pee + ee;
            alpha_s = al;
        }
        __syncthreads();

        const float al = alpha_s;
        creg[0] += al * hv0.x; creg[1] += al * hv0.y;
        creg[2] += al * hv0.z; creg[3] += al * hv0.w;
        creg[4] += al * hv1.x; creg[5] += al * hv1.y;
        creg[6] += al * hv1.z; creg[7] += al * hv1.w;
        // next iteration's alpha_s write happens after every thread has read
        // alpha_s, and is ordered by the first __syncthreads of that iter.
    }

    // Merge this chunk's partial c_t with hardware fp32 global atomics.
    float* __restrict__ cp = ct_out + (size_t)b * DD + d0;
    #pragma unroll
    for (int j = 0; j < 8; ++j) {
        __hip_atomic_fetch_add(&cp[j], creg[j],
                               __ATOMIC_RELAXED, __HIP_MEMORY_SCOPE_AGENT);
    }
}

// ---------------------------------------------------------------------------
// Launch
// ---------------------------------------------------------------------------
extern "C" void kernel_launch(void* const* d_in, const int* in_sizes, int n_in,
                              void* d_out, int out_size, void* d_ws, size_t ws_size,
                              hipStream_t stream)
{
    (void)in_sizes; (void)n_in; (void)out_size; (void)ws_size;

    // Inputs in setup_inputs() order
    const float* h_t     = (const float*)d_in[0];   // [B, D]
    const float* h       = (const float*)d_in[1];   // [B, N, D]
    const float* past_e  = (const float*)d_in[2];   // [B, N]
    const float* past_ee = (const float*)d_in[3];   // [B, N]
    const float* W_attn  = (const float*)d_in[4];   // [D, D]

    // Outputs concatenated in return order: alpha_t, c_t, new_past_e, new_past_ee
    float* out       = (float*)d_out;
    float* alpha_out = out;                               // B*N
    float* ct_out    = out + (size_t)BB * NN;             // B*D
    float* npe_out   = ct_out + (size_t)BB * DD;          // B*N
    float* npee_out  = npe_out + (size_t)BB * NN;         // B*N

    float* q = (float*)d_ws;                              // [B, D] scratch (1 MB)

    // 1) q = h_t @ W_attn via fp32 WMMA. 8 x 128 = 1024 wave-tiles,
    //    4 waves/block -> 256 blocks of 128 threads.
    qgemm_wmma_f32<<<256, 128, 0, stream>>>(h_t, W_attn, q);

    // 2) zero c_t accumulation slab
    {
        const int n = BB * DD;
        zero_ct<<<(n + 255) / 256, 256, 0, stream>>>(ct_out, n);
    }

    // 3) fused single pass over h
    fused_attn_pass<<<BB * NCHUNK, 256, 0, stream>>>(
        h, q, past_e, past_ee, alpha_out, ct_out, npe_out, npee_out);
}